// PhysicsInformedLoss_20246475833444
// MI455X (gfx1250) — compile-verified
//
#include <hip/hip_runtime.h>
#include <hip/hip_bf16.h>
#include <stdint.h>

#define RHO_L    1000.0f
#define RHO_G    1.0f
#define MU_C     0.001f
#define CP_C     4186.0f
#define DT_C     1e-05f
#define W_MASS   0.1f
#define W_MOM    0.05f
#define W_ENERGY 0.01f
#define EPS_C    1e-08f

#define TILE 256

#ifndef __has_builtin
#define __has_builtin(x) 0
#endif

// ---------------- CDNA5 async global->LDS path (ASYNCcnt) ------------------
#if defined(__AMDGCN__) && __has_builtin(__builtin_amdgcn_global_load_async_to_lds_b32) && \
    __has_builtin(__builtin_amdgcn_s_wait_asynccnt)
#define HAVE_ASYNC_LDS 1
#else
#define HAVE_ASYNC_LDS 0
#endif

#if HAVE_ASYNC_LDS
typedef __attribute__((address_space(1))) int as1_int;
typedef __attribute__((address_space(3))) int as3_int;
__device__ __forceinline__ void async_copy_b32(const void* g, void* l) {
  as1_int* gp = (as1_int*)(uintptr_t)g;                 // generic -> global (same bits)
  as3_int* lp = (as3_int*)(uint32_t)(uintptr_t)l;       // low 32 bits of generic LDS ptr == LDS offset
  __builtin_amdgcn_global_load_async_to_lds_b32(gp, lp, 0, 0);
}
#endif

// ---------------- CDNA5 WMMA wave-sum (V_WMMA_F32_16X16X4_F32) -------------
// D = A x B (+0) with B == ones, A-VGPR0 = per-lane value, A-VGPR1 = 0:
// every D[m][n] = row-sum of A, so the sum over one D column == sum of all A
// elements == wave-wide sum of the value. Column n is split lane n (rows 0-7)
// / lane n+16 (rows 8-15), so vector-sum the 8 accumulators then add the
// lane^16 partner. Result: all 32 lanes hold the wave total.
typedef __attribute__((ext_vector_type(2))) float v2f;
typedef __attribute__((ext_vector_type(8))) float v8f;

#if defined(__AMDGCN__) && __has_builtin(__builtin_amdgcn_wmma_f32_16x16x4_f32)
#define HAVE_WMMA 1
#else
#define HAVE_WMMA 0
#endif

__device__ __forceinline__ float wave_sum(float v) {
#if HAVE_WMMA
  v2f a; a.x = v;    a.y = 0.0f;
  v2f b; b.x = 1.0f; b.y = 1.0f;
  v8f c = {};
  c = __builtin_amdgcn_wmma_f32_16x16x4_f32(false, a, false, b, (short)0, c, false, false);
  float s = ((c[0] + c[1]) + (c[2] + c[3])) + ((c[4] + c[5]) + (c[6] + c[7]));
  s += __shfl_xor(s, 16, 32);
  return s;                      // all lanes hold the wave total
#else
#pragma unroll
  for (int off = 16; off > 0; off >>= 1) v += __shfl_down(v, off, 32);
  return v;                      // lane 0 holds the wave total
#endif
}

// ---------------------------------------------------------------------------
// Pass 1 over edges: degree count, divergence-of-momentum sum, grad(u) sum.
// ---------------------------------------------------------------------------
__global__ __launch_bounds__(TILE) void edge_pass1(
    const int* __restrict__ rowI, const int* __restrict__ colI,
    const float* __restrict__ pos, const float* __restrict__ x,
    float* __restrict__ cnt, float* __restrict__ divSum,
    float* __restrict__ gradSum, int E)
{
  __shared__ int sRow[TILE];
  __shared__ int sCol[TILE];
  const int tid = threadIdx.x;
  const int nTiles = (E + TILE - 1) / TILE;
  for (int t = blockIdx.x; t < nTiles; t += gridDim.x) {
    const int base = t * TILE;
    const int e = base + tid;
    // stream-ahead prefetch of the next tile this block will touch
    const int nb = base + (int)gridDim.x * TILE;
    __builtin_prefetch(rowI + nb + tid, 0, 1);
    __builtin_prefetch(colI + nb + tid, 0, 1);
    if (e < E) {
#if HAVE_ASYNC_LDS
      async_copy_b32(rowI + e, &sRow[tid]);
      async_copy_b32(colI + e, &sCol[tid]);
      __builtin_amdgcn_s_wait_asynccnt(0);
#else
      sRow[tid] = rowI[e];
      sCol[tid] = colI[e];
#endif
      const int r = sRow[tid];
      const int c = sCol[tid];
      const float prx = pos[3*(size_t)r+0], pry = pos[3*(size_t)r+1], prz = pos[3*(size_t)r+2];
      const float pcx = pos[3*(size_t)c+0], pcy = pos[3*(size_t)c+1], pcz = pos[3*(size_t)c+2];
      const float dxx = pcx - prx, dxy = pcy - pry, dxz = pcz - prz;
      const float inv = 1.0f / (dxx*dxx + dxy*dxy + dxz*dxz + EPS_C);
      // x row layout: [.,.,.,T,phase,vx,vy,vz]; load phase..vz as one b128
      const float4 xr = *reinterpret_cast<const float4*>(x + 8*(size_t)r + 4);
      const float4 xc = *reinterpret_cast<const float4*>(x + 8*(size_t)c + 4);
      const float rhor = (xr.x < 1.5f) ? RHO_L : RHO_G;
      const float rhoc = (xc.x < 1.5f) ? RHO_L : RHO_G;
      // divergence of momentum contribution
      const float dm0 = rhoc*xc.y - rhor*xr.y;
      const float dm1 = rhoc*xc.z - rhor*xr.z;
      const float dm2 = rhoc*xc.w - rhor*xr.w;
      const float divc = (dm0*dxx + dm1*dxy + dm2*dxz) * inv;
      // grad(u) contribution: g[i][j] = (du_i * inv) * dx_j
      const float g0 = (xc.y - xr.y) * inv;
      const float g1 = (xc.z - xr.z) * inv;
      const float g2 = (xc.w - xr.w) * inv;
      float* gs = gradSum + 9*(size_t)r;
      atomicAdd(&cnt[r], 1.0f);
      atomicAdd(&divSum[r], divc);
      atomicAdd(gs+0, g0*dxx); atomicAdd(gs+1, g0*dxy); atomicAdd(gs+2, g0*dxz);
      atomicAdd(gs+3, g1*dxx); atomicAdd(gs+4, g1*dxy); atomicAdd(gs+5, g1*dxz);
      atomicAdd(gs+6, g2*dxx); atomicAdd(gs+7, g2*dxy); atomicAdd(gs+8, g2*dxz);
    }
  }
}

// ---------------------------------------------------------------------------
// Normalize grad sums -> grad means (in place).
// ---------------------------------------------------------------------------
__global__ void normalize_grad(float* __restrict__ grad,
                               const float* __restrict__ cnt, int N)
{
  const int n = blockIdx.x * blockDim.x + threadIdx.x;
  if (n < N) {
    const float ic = 1.0f / fmaxf(cnt[n], 1.0f);
    float* g = grad + 9*(size_t)n;
#pragma unroll
    for (int k = 0; k < 9; ++k) g[k] *= ic;
  }
}

// ---------------------------------------------------------------------------
// Pass 2 over edges: Laplacian(u) sums from gathered grad(u).
// ---------------------------------------------------------------------------
__global__ __launch_bounds__(TILE) void edge_pass2(
    const int* __restrict__ rowI, const int* __restrict__ colI,
    const float* __restrict__ pos, const float* __restrict__ grad,
    float* __restrict__ lapSum, int E)
{
  __shared__ int sRow[TILE];
  __shared__ int sCol[TILE];
  const int tid = threadIdx.x;
  const int nTiles = (E + TILE - 1) / TILE;
  for (int t = blockIdx.x; t < nTiles; t += gridDim.x) {
    const int base = t * TILE;
    const int e = base + tid;
    const int nb = base + (int)gridDim.x * TILE;
    __builtin_prefetch(rowI + nb + tid, 0, 1);
    __builtin_prefetch(colI + nb + tid, 0, 1);
    if (e < E) {
#if HAVE_ASYNC_LDS
      async_copy_b32(rowI + e, &sRow[tid]);
      async_copy_b32(colI + e, &sCol[tid]);
      __builtin_amdgcn_s_wait_asynccnt(0);
#else
      sRow[tid] = rowI[e];
      sCol[tid] = colI[e];
#endif
      const int r = sRow[tid];
      const int c = sCol[tid];
      const float prx = pos[3*(size_t)r+0], pry = pos[3*(size_t)r+1], prz = pos[3*(size_t)r+2];
      const float pcx = pos[3*(size_t)c+0], pcy = pos[3*(size_t)c+1], pcz = pos[3*(size_t)c+2];
      const float dxx = pcx - prx, dxy = pcy - pry, dxz = pcz - prz;
      const float inv = 1.0f / (dxx*dxx + dxy*dxy + dxz*dxz + EPS_C);
      const float* gr = grad + 9*(size_t)r;
      const float* gc = grad + 9*(size_t)c;
      const float l0 = ((gc[0]-gr[0])*dxx + (gc[1]-gr[1])*dxy + (gc[2]-gr[2])*dxz) * inv;
      const float l1 = ((gc[3]-gr[3])*dxx + (gc[4]-gr[4])*dxy + (gc[5]-gr[5])*dxz) * inv;
      const float l2 = ((gc[6]-gr[6])*dxx + (gc[7]-gr[7])*dxy + (gc[8]-gr[8])*dxz) * inv;
      atomicAdd(&lapSum[3*(size_t)r+0], l0);
      atomicAdd(&lapSum[3*(size_t)r+1], l1);
      atomicAdd(&lapSum[3*(size_t)r+2], l2);
    }
  }
}

// ---------------------------------------------------------------------------
// Per-node loss terms + block reduction (WMMA wave-sum + LDS across waves).
// partial[3*block] = {sum mass_term, sum mom_term, sum energy_term}
// ---------------------------------------------------------------------------
__global__ __launch_bounds__(256) void node_loss(
    const float* __restrict__ pred, const float* __restrict__ x,
    const float* __restrict__ ef, const float* __restrict__ heat,
    const float* __restrict__ cnt, const float* __restrict__ divSum,
    const float* __restrict__ lapSum, float* __restrict__ partial, int N)
{
  const int n = blockIdx.x * blockDim.x + threadIdx.x;
  float massT = 0.f, momT = 0.f, enT = 0.f;
  if (n < N) {
    const float ic = 1.0f / fmaxf(cnt[n], 1.0f);
    const float dm = divSum[n] * ic;
    massT = dm * dm;

    const float du0 = pred[5*(size_t)n + 2];
    const float du1 = pred[5*(size_t)n + 3];
    const float du2 = pred[5*(size_t)n + 4];

    const float kVisc = (MU_C / RHO_L) * DT_C;   // applied to laplacian
    const float kForc = DT_C / RHO_L;            // applied to external force
    const float p0 = lapSum[3*(size_t)n+0] * ic * kVisc + ef[3*(size_t)n+0] * kForc;
    const float p1 = lapSum[3*(size_t)n+1] * ic * kVisc + ef[3*(size_t)n+1] * kForc;
    const float p2 = lapSum[3*(size_t)n+2] * ic * kVisc + ef[3*(size_t)n+2] * kForc;
    const float e0 = du0 - p0, e1 = du1 - p1, e2 = du2 - p2;
    momT = e0*e0 + e1*e1 + e2*e2;

    const float vx = x[8*(size_t)n + 5];
    const float vy = x[8*(size_t)n + 6];
    const float vz = x[8*(size_t)n + 7];
    const float keC = 0.5f * RHO_L * (vx*vx + vy*vy + vz*vz);
    const float wx = vx + du0, wy = vy + du1, wz = vz + du2;
    const float keN = 0.5f * RHO_L * (wx*wx + wy*wy + wz*wz);
    const float dE = (keN - keC) + RHO_L * CP_C * pred[5*(size_t)n + 0];
    const float de = dE - heat[n];
    enT = de * de;
  }

  // wave32 reduction on the matrix pipe (all lanes reconverged here)
  massT = wave_sum(massT);
  momT  = wave_sum(momT);
  enT   = wave_sum(enT);

  __shared__ float sred[3][8];
  const int lane = threadIdx.x & 31;
  const int wid  = threadIdx.x >> 5;
  if (lane == 0) { sred[0][wid] = massT; sred[1][wid] = momT; sred[2][wid] = enT; }
  __syncthreads();
  if (wid == 0) {                       // wave-uniform branch: wave 0 fully active
    const int nw = blockDim.x >> 5;
    float a = (lane < nw) ? sred[0][lane] : 0.f;
    float b = (lane < nw) ? sred[1][lane] : 0.f;
    float c = (lane < nw) ? sred[2][lane] : 0.f;
    a = wave_sum(a);
    b = wave_sum(b);
    c = wave_sum(c);
    if (lane == 0) {
      partial[3*(size_t)blockIdx.x + 0] = a;
      partial[3*(size_t)blockIdx.x + 1] = b;
      partial[3*(size_t)blockIdx.x + 2] = c;
    }
  }
}

// ---------------------------------------------------------------------------
// Deterministic final reduction over block partials -> scalar loss.
// ---------------------------------------------------------------------------
__global__ __launch_bounds__(256) void final_reduce(
    const float* __restrict__ partial, int nblocks, int N, float* __restrict__ out)
{
  float a = 0.f, b = 0.f, c = 0.f;
  for (int i = threadIdx.x; i < nblocks; i += blockDim.x) {
    a += partial[3*(size_t)i + 0];
    b += partial[3*(size_t)i + 1];
    c += partial[3*(size_t)i + 2];
  }
  a = wave_sum(a);
  b = wave_sum(b);
  c = wave_sum(c);

  __shared__ float sred[3][8];
  const int lane = threadIdx.x & 31;
  const int wid  = threadIdx.x >> 5;
  if (lane == 0) { sred[0][wid] = a; sred[1][wid] = b; sred[2][wid] = c; }
  __syncthreads();
  if (threadIdx.x == 0) {
    float A = 0.f, B = 0.f, C = 0.f;
    const int nw = blockDim.x >> 5;
    for (int w = 0; w < nw; ++w) { A += sred[0][w]; B += sred[1][w]; C += sred[2][w]; }
    const float invN = 1.0f / (float)N;
    out[0] = W_MASS * A * invN + W_MOM * B * (invN / 3.0f) + W_ENERGY * C * invN;
  }
}

// ---------------------------------------------------------------------------
// Launch: inputs per setup_inputs() order:
// 0 pred (N,5) f32 | 1 target (N,5) f32 | 2 x (N,8) f32 | 3 pos (N,3) f32
// 4 edge_index (2,E) int | 5 edge_attr (E,4) f32 | 6 external_force (N,3) f32
// 7 heat_source (N,) f32
// ---------------------------------------------------------------------------
extern "C" void kernel_launch(void* const* d_in, const int* in_sizes, int n_in,
                              void* d_out, int out_size, void* d_ws, size_t ws_size,
                              hipStream_t stream) {
  const float* pred = (const float*)d_in[0];
  const float* x    = (const float*)d_in[2];
  const float* pos  = (const float*)d_in[3];
  const int*   ei   = (const int*)d_in[4];
  const float* ef   = (const float*)d_in[6];
  const float* heat = (const float*)d_in[7];

  const int N = in_sizes[7];
  const int E = in_sizes[4] / 2;
  const int* rowI = ei;
  const int* colI = ei + E;

  // workspace layout (floats): cnt[N] | divSum[N] | grad[9N] | lapSum[3N] | partial[3*nodeBlocks]
  float* wsf     = (float*)d_ws;
  float* cnt     = wsf;
  float* divSum  = cnt + N;
  float* grad    = divSum + N;
  float* lapSum  = grad + 9*(size_t)N;
  float* partial = lapSum + 3*(size_t)N;

  (void)hipMemsetAsync(d_ws, 0, (size_t)14 * (size_t)N * sizeof(float), stream);

  const int nodeBlocks = (N + 255) / 256;
  const int nTiles = (E + TILE - 1) / TILE;
  const int edgeBlocks = nTiles < 2048 ? nTiles : 2048;

  edge_pass1<<<edgeBlocks, TILE, 0, stream>>>(rowI, colI, pos, x, cnt, divSum, grad, E);
  normalize_grad<<<nodeBlocks, 256, 0, stream>>>(grad, cnt, N);
  edge_pass2<<<edgeBlocks, TILE, 0, stream>>>(rowI, colI, pos, grad, lapSum, E);
  node_loss<<<nodeBlocks, 256, 0, stream>>>(pred, x, ef, heat, cnt, divSum, lapSum, partial, N);
  final_reduce<<<1, 256, 0, stream>>>(partial, nodeBlocks, N, (float*)d_out);
}